// GNNDecoder_8581344657810
// MI455X (gfx1250) — compile-verified
//
#include <hip/hip_runtime.h>

#define NNODES 50000
#define HDIM   256
#define ODIM   512

typedef float v2f __attribute__((ext_vector_type(2)));
typedef float v8f __attribute__((ext_vector_type(8)));

#define GF_ACCUM 1
#define GF_RELU  2

// ---------------------------------------------------------------- utilities
__global__ void zero_f32(float* __restrict__ p, long n) {
  long i = (long)blockIdx.x * blockDim.x + threadIdx.x;
  long s = (long)gridDim.x * blockDim.x;
  for (; i < n; i += s) p[i] = 0.0f;
}

__global__ void deg_count(const int* __restrict__ idx, float* __restrict__ deg, int E) {
  int i = blockIdx.x * blockDim.x + threadIdx.x;
  int s = gridDim.x * blockDim.x;
  for (; i < E; i += s) atomicAdd(&deg[idx[i]], 1.0f);
}

// deg -> clip(deg,1)^(-1/2)
__global__ void deg_finalize(float* __restrict__ deg, int n) {
  int i = blockIdx.x * blockDim.x + threadIdx.x;
  if (i < n) deg[i] = rsqrtf(fmaxf(deg[i], 1.0f));
}

// ------------------------------------------------- SpMM: agg[dst] += x[src]*outn[src]
// one wave32 per edge; each lane moves 8 contiguous floats (2 x float4 gather)
__global__ void __launch_bounds__(256)
spmm_scatter(const float* __restrict__ x, const float* __restrict__ outn,
             const int* __restrict__ src, const int* __restrict__ dst,
             float* __restrict__ agg, int E) {
  int lane = threadIdx.x & 31;
  int wave = blockIdx.x * (blockDim.x >> 5) + (threadIdx.x >> 5);
  int nw   = gridDim.x * (blockDim.x >> 5);
  for (int e = wave; e < E; e += nw) {
    int s = src[e], d = dst[e];
    float sc = outn[s];
    const float4* xs = (const float4*)(x + (size_t)s * HDIM);
    float* ad = agg + (size_t)d * HDIM + lane * 8;
    float4 a = xs[lane * 2];
    float4 b = xs[lane * 2 + 1];
    atomicAdd(ad + 0, a.x * sc);
    atomicAdd(ad + 1, a.y * sc);
    atomicAdd(ad + 2, a.z * sc);
    atomicAdd(ad + 3, a.w * sc);
    atomicAdd(ad + 4, b.x * sc);
    atomicAdd(ad + 5, b.y * sc);
    atomicAdd(ad + 6, b.z * sc);
    atomicAdd(ad + 7, b.w * sc);
  }
}

// ------------------------------------------------- WMMA GEMM: C = rowscale(A) @ W (+bias)(+C)(ReLU)
// A: [M,256] f32, W: [256,Ncol] f32 row-major. K fixed at 256 (=HDIM).
// block = 128 threads = 4 waves; wave w -> rows [blockM+16w,+16), cols [16*blockIdx.x,+16)
__global__ void __launch_bounds__(128)
gemm_wmma_f32(const float* __restrict__ A, const float* __restrict__ rowscale,
              const float* __restrict__ W, const float* __restrict__ bias,
              float* __restrict__ C, int M, int Ncol, int flags) {
  const int K = HDIM;
  // W panel staged as (k-pair, n, k&1) so each lane's B fragment is one b64 LDS read
  __shared__ float Bsh[K * 16];
  int tid = threadIdx.x;
  int wave = tid >> 5, lane = tid & 31;
  int n0 = blockIdx.x * 16;
  int rowBase = blockIdx.y * 64 + wave * 16;

  for (int i = tid; i < K * 16; i += 128) {
    int k = i >> 4, n = i & 15;
    Bsh[(k >> 1) * 32 + n * 2 + (k & 1)] = W[(size_t)k * Ncol + n0 + n];
  }
  __syncthreads();

  int mlane = lane & 15;          // M (for A) / N (for B,C) within tile
  int khalf = (lane >> 4) * 2;    // lanes 0-15: K=k,k+1 ; lanes 16-31: K=k+2,k+3
  int row  = rowBase + mlane;
  int rowL = row < M ? row : M - 1;               // clamp loads; EXEC stays uniform for WMMA
  float sc = rowscale ? rowscale[rowL] : 1.0f;
  const float* Arow = A + (size_t)rowL * K;

  v8f acc = {};
#pragma unroll 4
  for (int kk = 0; kk < K; kk += 4) {
    v2f a = *(const v2f*)(Arow + kk + khalf);
    a.x *= sc; a.y *= sc;
    v2f b = *(const v2f*)(Bsh + ((kk + khalf) >> 1) * 32 + mlane * 2);
    acc = __builtin_amdgcn_wmma_f32_16x16x4_f32(false, a, false, b, (short)0, acc,
                                                false, false);
  }

  // C/D layout: col = n0 + (lane&15); vgpr r -> row rowBase + r + 8*(lane>=16)
  int col  = n0 + mlane;
  int rOff = (lane >> 4) * 8;
  float bval = bias ? bias[col] : 0.0f;
#pragma unroll
  for (int r = 0; r < 8; ++r) {
    int rr = rowBase + rOff + r;
    if (rr < M) {
      size_t off = (size_t)rr * Ncol + col;
      float v = acc[r] + bval;
      if (flags & GF_ACCUM) v += C[off];
      if (flags & GF_RELU)  v = fmaxf(v, 0.0f);
      C[off] = v;
    }
  }
}

// ------------------------------------------------- BatchNorm (training-mode batch stats)
__global__ void __launch_bounds__(HDIM)
bn_stats(const float* __restrict__ h, float* __restrict__ sums,
         float* __restrict__ sumsq, int M) {
  int c = threadIdx.x;   // one column per thread -> coalesced row reads
  float s = 0.0f, q = 0.0f;
  for (int r = blockIdx.x; r < M; r += gridDim.x) {
    float v = h[(size_t)r * HDIM + c];
    s += v; q += v * v;
  }
  atomicAdd(&sums[c], s);
  atomicAdd(&sumsq[c], q);
}

__global__ void bn_apply(float* __restrict__ h, const float* __restrict__ sums,
                         const float* __restrict__ sumsq, const float* __restrict__ gamma,
                         const float* __restrict__ beta, int M) {
  long n = (long)M * HDIM;
  long i = (long)blockIdx.x * blockDim.x + threadIdx.x;
  long st = (long)gridDim.x * blockDim.x;
  float invM = 1.0f / (float)M;
  for (; i < n; i += st) {
    int c = (int)(i & (HDIM - 1));
    float mu  = sums[c] * invM;
    float var = sumsq[c] * invM - mu * mu;
    h[i] = gamma[c] * (h[i] - mu) * rsqrtf(var + 1e-5f) + beta[c];
  }
}

// ---------------------------------------------------------------- driver
extern "C" void kernel_launch(void* const* d_in, const int* in_sizes, int n_in,
                              void* d_out, int out_size, void* d_ws, size_t ws_size,
                              hipStream_t stream) {
  (void)n_in; (void)out_size; (void)ws_size;
  const float* x = (const float*)d_in[0];
  const int* src[3] = { (const int*)d_in[1], (const int*)d_in[3], (const int*)d_in[5] };
  const int* dst[3] = { (const int*)d_in[2], (const int*)d_in[4], (const int*)d_in[6] };
  const int  E[3]   = { in_sizes[1], in_sizes[3], in_sizes[5] };
  // setup_inputs() dict order: layer0 rel (W,b) x3 -> layer1 rel (W,b) x3 -> fc0 -> bn0 -> fc1
  const float* W0[3] = { (const float*)d_in[7],  (const float*)d_in[9],  (const float*)d_in[11] };
  const float* b0[3] = { (const float*)d_in[8],  (const float*)d_in[10], (const float*)d_in[12] };
  const float* W1[3] = { (const float*)d_in[13], (const float*)d_in[15], (const float*)d_in[17] };
  const float* b1[3] = { (const float*)d_in[14], (const float*)d_in[16], (const float*)d_in[18] };
  const float* fc0_W = (const float*)d_in[19];
  const float* fc0_b = (const float*)d_in[20];
  const float* bn_g  = (const float*)d_in[21];
  const float* bn_b  = (const float*)d_in[22];
  const float* fc1_W = (const float*)d_in[23];
  const float* fc1_b = (const float*)d_in[24];

  float* ws = (float*)d_ws;
  float* outn[3], *inn[3];
  for (int r = 0; r < 3; ++r) { outn[r] = ws + (2*r) * NNODES; inn[r] = ws + (2*r+1) * NNODES; }
  float* agg = ws + 6L * NNODES;
  float* h0  = agg + (long)NNODES * HDIM;   // layer sums (reused as layer-1 accumulator)
  float* h1  = h0  + (long)NNODES * HDIM;   // post fc0/relu/bn
  float* bns = h1  + (long)NNODES * HDIM;   // HDIM sums
  float* bnq = bns + HDIM;                  // HDIM sumsq

  const long NH = (long)NNODES * HDIM;
  dim3 gGemmH((HDIM + 15) / 16, (NNODES + 63) / 64);
  dim3 gGemmO((ODIM + 15) / 16, (NNODES + 63) / 64);

  // degrees: out_deg from src, in_deg from dst, per relation; then rsqrt(clip(.,1))
  zero_f32<<<2048, 256, 0, stream>>>(ws, 6L * NNODES);
  for (int r = 0; r < 3; ++r) {
    int blk = (E[r] + 255) / 256;
    deg_count<<<blk, 256, 0, stream>>>(src[r], outn[r], E[r]);
    deg_count<<<blk, 256, 0, stream>>>(dst[r], inn[r],  E[r]);
  }
  deg_finalize<<<(6 * NNODES + 255) / 256, 256, 0, stream>>>(ws, 6 * NNODES);

  // ---- layer 0: sum_r gconv(x) ----
  for (int r = 0; r < 3; ++r) {
    zero_f32<<<4096, 256, 0, stream>>>(agg, NH);
    spmm_scatter<<<(E[r] + 7) / 8, 256, 0, stream>>>(x, outn[r], src[r], dst[r], agg, E[r]);
    gemm_wmma_f32<<<gGemmH, 128, 0, stream>>>(agg, inn[r], W0[r], b0[r], h0,
                                              NNODES, HDIM, r == 0 ? 0 : GF_ACCUM);
  }
  // fc0 + relu
  gemm_wmma_f32<<<gGemmH, 128, 0, stream>>>(h0, nullptr, fc0_W, fc0_b, h1,
                                            NNODES, HDIM, GF_RELU);
  // batchnorm (batch stats, biased var)
  zero_f32<<<1, 256, 0, stream>>>(bns, 2L * HDIM);
  bn_stats<<<1024, HDIM, 0, stream>>>(h1, bns, bnq, NNODES);
  bn_apply<<<4096, 256, 0, stream>>>(h1, bns, bnq, bn_g, bn_b, NNODES);

  // ---- layer 1: sum_r gconv(h1) ----
  for (int r = 0; r < 3; ++r) {
    zero_f32<<<4096, 256, 0, stream>>>(agg, NH);
    spmm_scatter<<<(E[r] + 7) / 8, 256, 0, stream>>>(h1, outn[r], src[r], dst[r], agg, E[r]);
    gemm_wmma_f32<<<gGemmH, 128, 0, stream>>>(agg, inn[r], W1[r], b1[r], h0,
                                              NNODES, HDIM, r == 0 ? 0 : GF_ACCUM);
  }
  // fc1 -> d_out [N, 512]
  gemm_wmma_f32<<<gGemmO, 128, 0, stream>>>(h0, nullptr, fc1_W, fc1_b, (float*)d_out,
                                            NNODES, ODIM, 0);
}